// CSPN_6468220748336
// MI455X (gfx1250) — compile-verified
//
#include <hip/hip_runtime.h>

#define KK 7
#define K2 49
#define CENTER 24
#define H 512
#define W 512
#define GWD 518                 // guide plane H/W (512 + 6)
#define PLANE (518 * 518)
#define TILE 64
#define HALO 3
#define LW (TILE + 2 * HALO)    // 70
#define LELEMS (LW * LW)        // 4900
#define NTHREADS 256

typedef __attribute__((address_space(1))) int* gptr_t;
typedef __attribute__((address_space(3))) int* lptr_t;

// ---- CDNA5 async global -> LDS copy (ASYNCcnt path) -----------------------
__device__ __forceinline__ void async_copy_f32(const float* g, float* l) {
#if __has_builtin(__builtin_amdgcn_global_load_async_to_lds_b32)
  __builtin_amdgcn_global_load_async_to_lds_b32(
      (gptr_t)g, (lptr_t)l, /*offset=*/0, /*cpol=*/0);
#else
  unsigned lds_off =
      (unsigned)(__SIZE_TYPE__)(__attribute__((address_space(3))) char*)l;
  asm volatile("global_load_async_to_lds_b32 %0, %1, off"
               :: "v"(lds_off), "v"((unsigned long long)g)
               : "memory");
#endif
}

__device__ __forceinline__ void wait_async_zero() {
#if __has_builtin(__builtin_amdgcn_s_wait_asynccnt)
  __builtin_amdgcn_s_wait_asynccnt(0);
#else
  asm volatile("s_wait_asynccnt 0x0" ::: "memory");
#endif
}

// ---------------------------------------------------------------------------
__global__ __launch_bounds__(NTHREADS) void cspn_kernel(
    const float* __restrict__ gw,   // (B, 49, 518, 518)
    const float* __restrict__ hn,   // (B, 512, 512)
    const float* __restrict__ h0,   // (B, 512, 512)
    float* __restrict__ out)        // (B, 512, 512)
{
  __shared__ float tile[LELEMS];

  const int tid = threadIdx.x;
  const int tx  = tid & (TILE - 1);   // 0..63
  const int ty  = tid >> 6;           // 0..3
  const int bx  = blockIdx.x * TILE;
  const int by  = blockIdx.y * TILE;
  const int b   = blockIdx.z;

  const float* hnb = hn + (size_t)b * (H * W);

  // Stage hn tile (with halo, zero padded at image borders) into LDS.
  // In-range elements go through the async DMA path; border zeros via ds_store.
  for (int idx = tid; idx < LELEMS; idx += NTHREADS) {
    const int ly = idx / LW;
    const int lx = idx - ly * LW;
    const int gy = by - HALO + ly;
    const int gx = bx - HALO + lx;
    if ((unsigned)gy < (unsigned)H && (unsigned)gx < (unsigned)W) {
      async_copy_f32(hnb + (size_t)gy * W + gx, &tile[idx]);
    } else {
      tile[idx] = 0.0f;
    }
  }
  wait_async_zero();
  __syncthreads();

  const float* gwb  = gw + (size_t)b * K2 * PLANE;
  const float* h0b  = h0 + (size_t)b * (H * W);
  float*       outb = out + (size_t)b * (H * W);

  #pragma unroll 1
  for (int r = 0; r < TILE / 4; ++r) {
    const int yy = ty + 4 * r;                 // row inside tile, 0..63
    const int y  = by + yy;
    const int x  = bx + tx;

    // guide weights for this pixel: plane stride apart, streamed NT
    const float* gwp   = gwb + (size_t)(y + HALO) * GWD + (x + HALO);
    const float* lbase = &tile[yy * LW + tx];

    // center tap uses h0
    float acc = __builtin_nontemporal_load(gwp + (size_t)CENTER * PLANE) *
                h0b[(size_t)y * W + x];

    #pragma unroll
    for (int t = 0; t < K2; ++t) {
      if (t == CENTER) continue;
      const int i = t / KK;
      const int j = t - i * KK;
      const float wv = __builtin_nontemporal_load(gwp + (size_t)t * PLANE);
      const float hv = lbase[(6 - i) * LW + (6 - j)];   // imm-offset ds_load
      acc = fmaf(wv, hv, acc);
    }

    __builtin_nontemporal_store(acc, outb + (size_t)y * W + x);
  }
}

// ---------------------------------------------------------------------------
extern "C" void kernel_launch(void* const* d_in, const int* in_sizes, int n_in,
                              void* d_out, int out_size, void* d_ws, size_t ws_size,
                              hipStream_t stream) {
  const float* gw = (const float*)d_in[0];
  const float* hn = (const float*)d_in[1];
  const float* h0 = (const float*)d_in[2];
  float* out = (float*)d_out;

  const int B = in_sizes[1] / (H * W);   // hn is (B,1,H,W)

  dim3 grid(W / TILE, H / TILE, B);
  dim3 block(NTHREADS);
  hipLaunchKernelGGL(cspn_kernel, grid, block, 0, stream, gw, hn, h0, out);
}